// SimpleDescent_76605036692353
// MI455X (gfx1250) — compile-verified
//
#include <hip/hip_runtime.h>
#include <math.h>

// ---------------- problem constants ----------------
#define BB   16          // batch
#define DD   65536       // pixels per image (256*256)
#define HWW  256

static constexpr float GMM_VAR       = 0.01f;                 // sigma^2
static constexpr float TV_EPS        = 1e-7f;
static constexpr float LRATE         = 1e-2f;
static constexpr float BETA1         = 0.9f;
static constexpr float BETA2         = 0.999f;
static constexpr float ADAM_EPS      = 1e-8f;
static constexpr float REG_W         = 0.1f;
static constexpr float INV_SIG_SQRT2 = 7.0710678118654755f;   // 1/(sigma*sqrt(2))
static constexpr float INV_2VAR      = 50.0f;                 // 1/(2 sigma^2)
static constexpr float PDF_COEF      = 3.989422804014327f;    // 1/(sigma*sqrt(2 pi))

typedef __attribute__((ext_vector_type(2))) float v2f;
typedef __attribute__((ext_vector_type(8))) float v8f;

// ---------------- init / zero ----------------
__global__ void k_init_zero(float* __restrict__ p, size_t n) {
    size_t i = (size_t)blockIdx.x * blockDim.x + threadIdx.x;
    size_t stride = (size_t)gridDim.x * blockDim.x;
    for (; i < n; i += stride) p[i] = 0.0f;
}

__global__ void k_zero_small(float* __restrict__ mean_acc, float* __restrict__ C) {
    int t = threadIdx.x;
    if (t < 16) mean_acc[t] = 0.0f;
    C[t] = 0.0f;   // 256 threads zero 16x16
}

// ---------------- residual + per-image sum ----------------
__global__ void __launch_bounds__(256)
k_resid(const float* __restrict__ u, const float* __restrict__ f,
        float* __restrict__ z, float* __restrict__ mean_acc) {
    const int b = blockIdx.y;
    const int p = blockIdx.x * 256 + threadIdx.x;      // grid.x = 256 -> covers 65536
    const size_t idx = (size_t)b * DD + p;
    const float zv = u[idx] - f[idx];
    z[idx] = zv;
    __shared__ float red[256];
    red[threadIdx.x] = zv;
    __syncthreads();
    for (int s = 128; s > 0; s >>= 1) {
        if (threadIdx.x < s) red[threadIdx.x] += red[threadIdx.x + s];
        __syncthreads();
    }
    if (threadIdx.x == 0) atomicAdd(&mean_acc[b], red[0]);
}

// ---------------- full-image bitonic sort in 256KB LDS ----------------
// CDNA5: 320KB LDS per WGP lets one workgroup hold an entire 256x256 fp32
// image. One workgroup per image, 1024 threads (32 wave32 waves).
__global__ void __launch_bounds__(1024)
k_sort(const float* __restrict__ z, float* __restrict__ zs) {
    extern __shared__ float s[];
    const int b   = blockIdx.x;
    const int tid = threadIdx.x;
    const float* zb = z + (size_t)b * DD;
    for (int i = tid; i < DD; i += 1024) s[i] = zb[i];
    __syncthreads();
    for (int k = 2; k <= DD; k <<= 1) {
        for (int j = k >> 1; j > 0; j >>= 1) {
            for (int i = tid; i < DD; i += 1024) {
                const int ixj = i ^ j;
                if (ixj > i) {
                    const float a = s[i], c = s[ixj];
                    const bool swap = ((i & k) == 0) ? (a > c) : (a < c);
                    if (swap) { s[i] = c; s[ixj] = a; }
                }
            }
            __syncthreads();
        }
    }
    float* out = zs + (size_t)b * DD;
    for (int i = tid; i < DD; i += 1024) out[i] = s[i];
}

// ---------------- covariance GEMM via fp32 WMMA ----------------
// C[i][j] += sum_p zc[i][p]*zc[j][p]   (16x16x65536 fp32 GEMM)
// V_WMMA_F32_16X16X4_F32: A is 16x4 (lanes 0-15: M=lane, K={0,1}; lanes
// 16-31: M=lane-16, K={2,3}), B is 4x16. Because the product is symmetric
// (B[k][n] = zc[n][p0+k] = A[n][k]), the same per-lane register pair serves
// as both A and B operands.
__global__ void __launch_bounds__(128)
k_cov(const float* __restrict__ z, const float* __restrict__ mean_acc,
      float* __restrict__ C) {
    __shared__ float tile[16][65];   // pad to 65 to dodge bank conflicts
    __shared__ float smean[16];
    const int t = threadIdx.x;
    if (t < 16) smean[t] = mean_acc[t] * (1.0f / (float)DD);
    __syncthreads();

    const int lane  = t & 31;
    const int wave  = t >> 5;            // 4 waves per block
    const int mrow  = lane & 15;
    const int kbase = (lane < 16) ? 0 : 2;

    v8f acc = {};
    const int ntiles = DD / 64;
    for (int tileIdx = blockIdx.x; tileIdx < ntiles; tileIdx += gridDim.x) {
        const int p0  = tileIdx * 64;
        const int col = t & 63, row0 = t >> 6;     // 128 thr -> 2 rows/pass
        #pragma unroll
        for (int r = 0; r < 8; ++r) {
            const int row = row0 + r * 2;
            tile[row][col] = z[(size_t)row * DD + p0 + col] - smean[row];
        }
        // prefetch next tile (gfx1250 global_prefetch_b8) while WMMAs run
        const int pnext = p0 + 64 * gridDim.x;
        if (pnext < DD) {
            #pragma unroll
            for (int r = 0; r < 8; ++r)
                __builtin_prefetch(&z[(size_t)(row0 + r * 2) * DD + pnext + col], 0, 3);
        }
        __syncthreads();
        const int base = wave * 16;                // 16 pixels per wave
        #pragma unroll
        for (int c4 = 0; c4 < 4; ++c4) {
            const int pp = base + c4 * 4 + kbase;
            v2f a;
            a.x = tile[mrow][pp];
            a.y = tile[mrow][pp + 1];
            acc = __builtin_amdgcn_wmma_f32_16x16x4_f32(
                false, a, false, a, (short)0, acc, false, false);
        }
        __syncthreads();
    }
    // C/D layout: VGPR r, lanes 0-15 -> (M=r, N=lane); lanes 16-31 -> (M=r+8)
    const int ncol = lane & 15;
    #pragma unroll
    for (int r = 0; r < 8; ++r) {
        const int mrowD = (lane < 16) ? r : (r + 8);
        atomicAdd(&C[mrowD * 16 + ncol], acc[r]);
    }
}

// ---------------- fused gradient + Adam ----------------
__global__ void __launch_bounds__(256)
k_grad(const float* __restrict__ u, const float* __restrict__ f,
       const float* __restrict__ z, const float* __restrict__ zs,
       const float* __restrict__ mean_acc, const float* __restrict__ Cacc,
       float* __restrict__ m, float* __restrict__ v,
       float* __restrict__ u_new, float bc1, float bc2) {
    __shared__ float sE[16][16];
    __shared__ float smean[16];
    const int t = threadIdx.x;
    const int b = blockIdx.y;
    if (t < 16) smean[t] = mean_acc[t] * (1.0f / (float)DD);
    {
        const int i = t >> 4, j = t & 15;
        const float c = Cacc[t] * (1.0f / (float)DD);
        sE[i][j] = c - ((i == j) ? GMM_VAR : 0.0f);
    }
    __syncthreads();

    const int p = blockIdx.x * 256 + t;
    const size_t idx = (size_t)b * DD + p;
    const float zv = z[idx];

    // --- KS term: rank via binary search in the sorted residuals ---
    const float* zsb = zs + (size_t)b * DD;
    int lo = 0, hi = DD;
    while (lo < hi) { const int mid = (lo + hi) >> 1; if (zsb[mid] < zv) lo = mid + 1; else hi = mid; }
    const float emp  = (float)(lo + 1) * (1.0f / (float)DD);
    const float cdf  = 0.5f * (1.0f + erff(zv * INV_SIG_SQRT2));
    const float diff = emp - cdf;
    const float sgn  = (diff > 0.f) ? 1.f : ((diff < 0.f) ? -1.f : 0.f);
    const float pdf  = PDF_COEF * __expf(-zv * zv * INV_2VAR);
    const float g_ks = -(1.0f / ((float)BB * (float)DD)) * sgn * pdf;

    // --- image fidelity ---
    const float g_img = (2.0f / ((float)BB * (float)DD)) * zv;

    // --- covariance: g = (4/(B^2 D)) * E * zc  (centering backprop cancels) ---
    float accv = 0.f;
    #pragma unroll
    for (int j = 0; j < 16; ++j)
        accv += sE[b][j] * (z[(size_t)j * DD + p] - smean[j]);
    const float g_cov = (4.0f / ((float)BB * (float)BB * (float)DD)) * accv;

    // --- isotropic TV divergence stencil ---
    const int y = p >> 8, x = p & 255;
    const float* ub = u + (size_t)b * DD;
    auto U = [&](int yy, int xx) -> float {
        return (yy >= 0 && yy < HWW && xx >= 0 && xx < HWW) ? ub[yy * HWW + xx] : 0.0f;
    };
    const float uc = U(y, x);
    const float d0 = uc - U(y, x - 1);
    const float d1 = uc - U(y - 1, x);
    float g = (d0 + d1) / sqrtf(d0 * d0 + d1 * d1 + TV_EPS);
    if (x + 1 < HWW) {
        const float ur = U(y, x + 1);
        const float e0 = ur - uc, e1 = ur - U(y - 1, x + 1);
        g -= e0 / sqrtf(e0 * e0 + e1 * e1 + TV_EPS);
    }
    if (y + 1 < HWW) {
        const float ud = U(y + 1, x);
        const float e0 = ud - U(y + 1, x - 1), e1 = ud - uc;
        g -= e1 / sqrtf(e0 * e0 + e1 * e1 + TV_EPS);
    }
    const float g_tv = (REG_W / ((float)BB * (float)DD)) * g;

    // --- Adam update ---
    const float gt = g_ks + g_img + g_cov + g_tv;
    const float mv = BETA1 * m[idx] + (1.0f - BETA1) * gt;
    const float vv = BETA2 * v[idx] + (1.0f - BETA2) * gt * gt;
    m[idx] = mv; v[idx] = vv;
    const float mh = mv / bc1;
    const float vh = vv / bc2;
    u_new[idx] = uc - LRATE * mh / (sqrtf(vh) + ADAM_EPS);
}

// ---------------- host driver ----------------
extern "C" void kernel_launch(void* const* d_in, const int* in_sizes, int n_in,
                              void* d_out, int out_size, void* d_ws, size_t ws_size,
                              hipStream_t stream) {
    (void)in_sizes; (void)n_in; (void)out_size; (void)ws_size;
    const float* f = (const float*)d_in[0];
    float* uA = (float*)d_out;                       // ping (holds u at even t; final)
    float* ws = (float*)d_ws;
    const size_t N = (size_t)BB * DD;
    float* uB       = ws;                            // pong
    float* mBuf     = uB + N;
    float* vBuf     = mBuf + N;
    float* zBuf     = vBuf + N;
    float* zsBuf    = zBuf + N;
    float* mean_acc = zsBuf + N;                     // 16
    float* Cacc     = mean_acc + 16;                 // 256

    // allow 256KB dynamic LDS for the full-image sort (CDNA5: 320KB/WGP)
    hipFuncSetAttribute(reinterpret_cast<const void*>(k_sort),
                        hipFuncAttributeMaxDynamicSharedMemorySize, DD * 4);

    // u0 = 0 (in d_out), m = v = 0
    k_init_zero<<<1024, 256, 0, stream>>>(uA, N);
    k_init_zero<<<2048, 256, 0, stream>>>(mBuf, 2 * N);  // m and v contiguous

    for (int t = 1; t <= 32; ++t) {
        float* u_old = (t & 1) ? uA : uB;            // t=1 reads zeros from d_out
        float* u_new = (t & 1) ? uB : uA;            // t=32 (even) writes d_out
        k_zero_small<<<1, 256, 0, stream>>>(mean_acc, Cacc);
        k_resid<<<dim3(256, BB), 256, 0, stream>>>(u_old, f, zBuf, mean_acc);
        k_sort<<<BB, 1024, DD * 4, stream>>>(zBuf, zsBuf);
        k_cov<<<16, 128, 0, stream>>>(zBuf, mean_acc, Cacc);
        const float bc1 = 1.0f - powf(BETA1, (float)t);
        const float bc2 = 1.0f - powf(BETA2, (float)t);
        k_grad<<<dim3(256, BB), 256, 0, stream>>>(u_old, f, zBuf, zsBuf,
                                                  mean_acc, Cacc, mBuf, vBuf,
                                                  u_new, bc1, bc2);
    }
}